// SpatialTransformerBspline_13503377179123
// MI455X (gfx1250) — compile-verified
//
#include <hip/hip_runtime.h>

// ---------------------------------------------------------------------------
// B-spline spatial transformer, MI455X (gfx1250, wave32).
//
// delta = Wv(512x74) * theta(74x74) * Wu^T(74x512)  (exact dense factorization
// of the separable cubic B-spline; Wv==Wu since H==W and both use sx=sy=512/71).
// Stage A: S[m][j] = sum_c Bv[c](h)*theta[py+c][j]  -> LDS (4-tap, cheap VALU)
// Stage B: 16x16 delta tile = S(16x80) x Wmat^T(80x16) via v_wmma_f32_16x16x4_f32
//          (K padded 74->80; B-fragment synthesized from the 4 analytic basis
//           weights per output column, no memory traffic for B).
// Epilogue: fused clamped bilinear gather of the NHWC fmap + stores.
// ---------------------------------------------------------------------------

typedef __attribute__((ext_vector_type(2))) float v2f;
typedef __attribute__((ext_vector_type(8))) float v8f;

#define HH   512
#define WW   512
#define CCH  3
#define NY   74
#define NX   74
#define KPAD 80   // K padded to multiple of 4 (and 16)
#define SROW 84   // LDS row stride (floats) -> conflict-free half-wave access

__device__ __forceinline__ float bsel(int d, float b0, float b1, float b2, float b3) {
    float r = 0.0f;
    r = (d == 0) ? b0 : r;
    r = (d == 1) ? b1 : r;
    r = (d == 2) ? b2 : r;
    r = (d == 3) ? b3 : r;
    return r;
}

__global__ __launch_bounds__(256) void stn_bspline_kernel(
    const float* __restrict__ fmap,   // [16,512,512,3]
    const float* __restrict__ theta,  // [16,2,74,74]
    float* __restrict__ out)          // [16,512,512,3] ++ [2,16,512,512]
{
    __shared__ float Sx[16][SROW];
    __shared__ float Sy[16][SROW];

    const float s  = 512.0f / 71.0f;          // sx == sy (reference uses W/gy)
    const int   b  = blockIdx.y;
    const int   h0 = blockIdx.x * 16;

    const float* thx = theta + (size_t)b * 2 * NY * NX;
    const float* thy = thx + NY * NX;

    // ---- Stage A: S[m][j] = sum_c Bv[c](h0+m) * theta[py(h0+m)+c][j] -------
    for (int idx = threadIdx.x; idx < 16 * KPAD; idx += 256) {
        const int m = idx / KPAD;
        const int j = idx - m * KPAD;
        float vx = 0.0f, vy = 0.0f;
        if (j < NX) {
            const float hf = (float)(h0 + m);
            const float q  = hf / s;
            const float pf = floorf(q);
            const float v  = q - pf;
            const int   py = (int)pf;          // py+3 <= 73 always
            const float v2 = v * v, v3 = v2 * v;
            const float c0 = (-v3 + 3.0f * v2 - 3.0f * v + 1.0f) * (1.0f / 6.0f);
            const float c1 = (3.0f * v3 - 6.0f * v2 + 4.0f) * (1.0f / 6.0f);
            const float c2 = (-3.0f * v3 + 3.0f * v2 + 3.0f * v + 1.0f) * (1.0f / 6.0f);
            const float c3 = v3 * (1.0f / 6.0f);
            const float* px_ = thx + (size_t)py * NX + j;
            const float* py_ = thy + (size_t)py * NX + j;
            vx = c0 * px_[0] + c1 * px_[NX] + c2 * px_[2 * NX] + c3 * px_[3 * NX];
            vy = c0 * py_[0] + c1 * py_[NX] + c2 * py_[2 * NX] + c3 * py_[3 * NX];
        }
        Sx[m][j] = vx;
        Sy[m][j] = vy;
    }
    __syncthreads();

    const int lane = threadIdx.x & 31;
    const int wave = threadIdx.x >> 5;
    const int n    = lane & 15;        // output column within tile / A row m
    const int half = lane >> 4;        // 0: K 0..1, 1: K 2..3
    const int koff = half << 1;

    const float* fb  = fmap + (size_t)b * HH * WW * CCH;
    float*       dlt = out + (size_t)16 * HH * WW * CCH;   // delta section

    for (int it = 0; it < 4; ++it) {   // 8 waves x 4 iters = 32 w-tiles
        const int w0 = (wave + 8 * it) * 16;
        const int wl = w0 + n;

        // Analytic Wmat row for output column wl: 4 nonzeros bu0..bu3 at p..p+3
        const float qx = (float)wl / s;
        const float pf = floorf(qx);
        const float u  = qx - pf;
        const int   p  = (int)pf;
        const float u2 = u * u, u3 = u2 * u;
        const float bu0 = (-u3 + 3.0f * u2 - 3.0f * u + 1.0f) * (1.0f / 6.0f);
        const float bu1 = (3.0f * u3 - 6.0f * u2 + 4.0f) * (1.0f / 6.0f);
        const float bu2 = (-3.0f * u3 + 3.0f * u2 + 3.0f * u + 1.0f) * (1.0f / 6.0f);
        const float bu3 = u3 * (1.0f / 6.0f);

        v8f cx = {0.0f, 0.0f, 0.0f, 0.0f, 0.0f, 0.0f, 0.0f, 0.0f};
        v8f cy = {0.0f, 0.0f, 0.0f, 0.0f, 0.0f, 0.0f, 0.0f, 0.0f};

        // ---- Stage B: delta tile = S(16x80) x Wmat^T(80x16) ----------------
#pragma unroll
        for (int j0 = 0; j0 < KPAD; j0 += 4) {
            const int k0 = j0 + koff;
            v2f ax = *(const v2f*)&Sx[n][k0];
            v2f ay = *(const v2f*)&Sy[n][k0];
            v2f bb;
            bb.x = bsel(k0 - p, bu0, bu1, bu2, bu3);
            bb.y = bsel(k0 + 1 - p, bu0, bu1, bu2, bu3);
            cx = __builtin_amdgcn_wmma_f32_16x16x4_f32(
                false, ax, false, bb, (short)0, cx, false, false);
            cy = __builtin_amdgcn_wmma_f32_16x16x4_f32(
                false, ay, false, bb, (short)0, cy, false, false);
        }

        // ---- Fused epilogue: clamped bilinear gather + stores --------------
#pragma unroll
        for (int i = 0; i < 8; ++i) {
            const int m  = i + (half << 3);    // C-layout: row = i + 8*(lane>=16)
            const int hh = h0 + m;
            const float dx = cx[i];
            const float dy = cy[i];
            const float xs = (float)wl + dx;
            const float ys = (float)hh + dy;
            const int x0 = (int)floorf(xs);
            const int y0 = (int)floorf(ys);
            const int x1c = min(max(x0 + 1, 0), WW - 1);
            const int y1c = min(max(y0 + 1, 0), HH - 1);
            const int x0c = min(max(x0, 0), WW - 1);
            const int y0c = min(max(y0, 0), HH - 1);
            const float x0f = (float)x0c, x1f = (float)x1c;
            const float y0f = (float)y0c, y1f = (float)y1c;
            const float wa = (x1f - xs) * (y1f - ys);
            const float wb = (x1f - xs) * (ys - y0f);
            const float wc = (xs - x0f) * (y1f - ys);
            const float wd = (xs - x0f) * (ys - y0f);
            const float* Ia = fb + (size_t)(y0c * WW + x0c) * CCH;
            const float* Ib = fb + (size_t)(y1c * WW + x0c) * CCH;
            const float* Ic = fb + (size_t)(y0c * WW + x1c) * CCH;
            const float* Id = fb + (size_t)(y1c * WW + x1c) * CCH;
            const size_t pix = ((size_t)b * HH + hh) * WW + wl;
            float* op = out + pix * CCH;
            op[0] = wa * Ia[0] + wb * Ib[0] + wc * Ic[0] + wd * Id[0];
            op[1] = wa * Ia[1] + wb * Ib[1] + wc * Ic[1] + wd * Id[1];
            op[2] = wa * Ia[2] + wb * Ib[2] + wc * Ic[2] + wd * Id[2];
            dlt[pix] = dx;
            dlt[(size_t)16 * HH * WW + pix] = dy;
        }
    }
}

extern "C" void kernel_launch(void* const* d_in, const int* in_sizes, int n_in,
                              void* d_out, int out_size, void* d_ws, size_t ws_size,
                              hipStream_t stream) {
    (void)in_sizes; (void)n_in; (void)out_size; (void)d_ws; (void)ws_size;
    const float* fmap  = (const float*)d_in[0];   // [16,512,512,3] f32
    const float* theta = (const float*)d_in[1];   // [16,2,74,74]  f32
    float* out = (float*)d_out;                   // out ++ delta, f32

    dim3 grid(32, 16);   // 32 h-tiles x 16 batches
    stn_bspline_kernel<<<grid, 256, 0, stream>>>(fmap, theta, out);
}